// LinearResizeLayer_16904991277188
// MI455X (gfx1250) — compile-verified
//
#include <hip/hip_runtime.h>
#include <stdint.h>

// Trilinear 2x upsample (2,64,64,64,32)f32 -> (2,128,128,128,32)f32.
// Legacy mapping src = dst/2  =>  lo = d>>1, frac = (d&1)*0.5, hi = min(lo+1,63).
// Separable linear passes commute, so each output is a trilinear blend of the
// 8 input corners of its 2x2x2 cell. Memory-bound: ~576 MB HBM => ~25 us floor.
// CDNA5 path used: GLOBAL_LOAD_ASYNC_TO_LDS_B128 staging + s_wait_asynccnt,
// non-temporal streaming stores to keep the 64 MB input resident in L2.

typedef float f4  __attribute__((ext_vector_type(4)));
typedef int   v4i __attribute__((ext_vector_type(4)));

#define ASG __attribute__((address_space(1)))
#define ASL __attribute__((address_space(3)))

__global__ __launch_bounds__(256) void resize2x3d_kernel(
    const float* __restrict__ in, float* __restrict__ out)
{
#if defined(__HIP_DEVICE_COMPILE__)
    const int tid = threadIdx.x;
    const int blk = (int)blockIdx.x;
    const int j = blk & 63;          // input y cell
    const int i = (blk >> 6) & 63;   // input x cell
    const int b = blk >> 12;         // batch

    // [row r = xi*2+yi][z 0..63][c4 0..7] of float4  == 32 KB
    __shared__ f4 lds4[4 * 64 * 8];

    // ---- stage the 4 needed input rows (each contiguous 8 KB) into LDS ----
    #pragma unroll
    for (int r = 0; r < 4; ++r) {
        const int xi = r >> 1, yi = r & 1;
        const int xr = (i + xi < 64) ? (i + xi) : 63;
        const int yr = (j + yi < 64) ? (j + yi) : 63;
        const f4* rb = (const f4*)(in + (size_t)((b * 64 + xr) * 64 + yr) * 2048);
        #pragma unroll
        for (int h = 0; h < 2; ++h) {
            const int off = tid + h * 256;            // [0,512) float4s per row
            const f4* g = rb + off;
            f4* l = &lds4[r * 512 + off];
#if __has_builtin(__builtin_amdgcn_global_load_async_to_lds_b128)
            __builtin_amdgcn_global_load_async_to_lds_b128(
                (ASG v4i*)(uintptr_t)g,
                (ASL v4i*)(ASL void*)(void*)l,
                /*offset=*/0, /*cpol=*/0);
#else
            asm volatile("global_load_async_to_lds_b128 %0, %1, off"
                         :: "v"((ASL void*)l), "v"(g)
                         : "memory");
#endif
        }
    }
#if __has_builtin(__builtin_amdgcn_s_wait_asynccnt)
    __builtin_amdgcn_s_wait_asynccnt(0);
#else
    asm volatile("s_wait_asynccnt 0" ::: "memory");
#endif
    __syncthreads();

    // ---- compute: 2048 z-pair cells per block, 8 per thread ----
    // cell n: c4 = n&7, k = (n>>3)&63, dy = (n>>9)&1, dx = (n>>10)&1
    #pragma unroll
    for (int it = 0; it < 8; ++it) {
        const int n  = tid + it * 256;
        const int c4 = n & 7;
        const int k  = (n >> 3) & 63;
        const int dy = (n >> 9) & 1;
        const int dx = (n >> 10) & 1;

        const int zh = (k + 1 < 64) ? (k + 1) : 63;

        const float fx  = dx ? 0.5f : 0.0f;
        const float fy  = dy ? 0.5f : 0.0f;
        const float w00 = (1.0f - fx) * (1.0f - fy);
        const float w01 = (1.0f - fx) * fy;
        const float w10 = fx * (1.0f - fy);
        const float w11 = fx * fy;

        const int zb  = k  * 8 + c4;
        const int zb2 = zh * 8 + c4;

        // xy-blend at z = k
        f4 a = w00 * lds4[zb]
             + w01 * lds4[512  + zb]
             + w10 * lds4[1024 + zb]
             + w11 * lds4[1536 + zb];
        // xy-blend at z = min(k+1,63)
        f4 bb = w00 * lds4[zb2]
              + w01 * lds4[512  + zb2]
              + w10 * lds4[1024 + zb2]
              + w11 * lds4[1536 + zb2];

        const f4 o0 = a;                    // oz = 2k   (fz = 0)
        const f4 o1 = 0.5f * (a + bb);      // oz = 2k+1 (fz = 0.5)

        const int ox = 2 * i + dx;
        const int oy = 2 * j + dy;
        f4* op = (f4*)(out +
            ((((size_t)(b * 128 + ox) * 128 + oy) * 128 + (size_t)(2 * k)) * 32
             + (size_t)(c4 * 4)));
        __builtin_nontemporal_store(o0, op);
        __builtin_nontemporal_store(o1, op + 8);   // next z => +32 floats
    }
#endif // __HIP_DEVICE_COMPILE__
}

extern "C" void kernel_launch(void* const* d_in, const int* in_sizes, int n_in,
                              void* d_out, int out_size, void* d_ws, size_t ws_size,
                              hipStream_t stream) {
    (void)in_sizes; (void)n_in; (void)out_size; (void)d_ws; (void)ws_size;
    const float* in = (const float*)d_in[0];
    float* out = (float*)d_out;
    dim3 grid(2 * 64 * 64);   // (b, i, j) cells
    dim3 block(256);
    hipLaunchKernelGGL(resize2x3d_kernel, grid, block, 0, stream, in, out);
}